// GAT_82497731822010
// MI455X (gfx1250) — compile-verified
//
#include <hip/hip_runtime.h>
#include <hip/hip_bf16.h>

typedef __attribute__((ext_vector_type(2))) float v2f;
typedef __attribute__((ext_vector_type(8))) float v8f;

#define NEG_SLOPE 0.2f
#define GAT_EPS 1e-16f
// order-preserving encode of -inf: ~0xFF800000
#define ENC_NEG_INF 0x007FFFFFu

__device__ __forceinline__ float lrelu(float e) { return e > 0.0f ? e : NEG_SLOPE * e; }

// Monotonic (order-preserving) float<->uint encoding so atomicMax(u32) == float max.
__device__ __forceinline__ unsigned fenc(float f) {
    unsigned u = __float_as_uint(f);
    return (u & 0x80000000u) ? ~u : (u | 0x80000000u);
}
__device__ __forceinline__ float fdec(unsigned u) {
    return __uint_as_float((u & 0x80000000u) ? (u ^ 0x80000000u) : ~u);
}

// ---------------------------------------------------------------------------
// init: out1 = broadcast(b1), out = broadcast(b2), m* = enc(-inf), z* = 0
// ---------------------------------------------------------------------------
__global__ void gat_init(float* __restrict__ out1, unsigned* __restrict__ m1,
                         float* __restrict__ z1, float* __restrict__ out,
                         unsigned* __restrict__ m2, float* __restrict__ z2,
                         const float* __restrict__ b1, const float* __restrict__ b2,
                         int N) {
    int tid = blockIdx.x * blockDim.x + threadIdx.x;
    int total = N * 128;
    if (tid < total) out1[tid] = b1[tid & 127];
    if (tid < N * 8) { m1[tid] = ENC_NEG_INF; z1[tid] = 0.0f; }
    if (tid < N * 3) out[tid] = b2[tid % 3];
    if (tid < N) { m2[tid] = ENC_NEG_INF; z2[tid] = 0.0f; }
}

// ---------------------------------------------------------------------------
// Layer-1 GEMM: h[N,128] = x[N,16] @ W1[16,128] via V_WMMA_F32_16X16X4_F32.
// One wave per 16x16 tile; 4 K-steps. Loads are index-clamped (unconditional);
// stores use a wave-uniform full-tile fast path with constant offsets.
// A 16x4 layout: lanes 0-15 M=lane (K0->v0,K1->v1); lanes 16-31 (K2->v0,K3->v1)
// B 4x16 layout: mirrors A in K; N = lane&15
// C 16x16:       VGPR r -> row r (lanes 0-15) / row r+8 (lanes 16-31), col = lane&15
// ---------------------------------------------------------------------------
__global__ void gemm1_wmma(const float* __restrict__ x, const float* __restrict__ W,
                           float* __restrict__ h, int N, int nRowTiles) {
    int gid  = blockIdx.x * blockDim.x + threadIdx.x;
    int wave = gid >> 5;
    int lane = threadIdx.x & 31;
    int rowTile = wave >> 3;   // 8 column tiles (128/16)
    int colTile = wave & 7;
    if (rowTile >= nRowTiles) return;           // wave-uniform exit
    int m0 = rowTile * 16;
    int j0 = colTile * 16;
    int mlane = m0 + (lane & 15);
    mlane = mlane < N ? mlane : 0;              // clamp: padded lanes reload row 0
    int klo   = (lane >> 4) << 1;               // 0 or 2
    int jlane = j0 + (lane & 15);
    const float* arow = x + (size_t)mlane * 16 + klo;  // 8B-aligned (klo even)
    v8f c = {};
#pragma unroll
    for (int k = 0; k < 16; k += 4) {
        v2f a = *(const v2f*)(arow + k);        // one global_load_b64
        v2f b;
        b.x = W[(k + klo) * 128 + jlane];
        b.y = W[(k + klo + 1) * 128 + jlane];
        c = __builtin_amdgcn_wmma_f32_16x16x4_f32(false, a, false, b, (short)0, c,
                                                  false, false);
    }
    int rbase = m0 + ((lane >> 4) << 3);
    float* hp = h + (size_t)rbase * 128 + jlane;
    if (m0 + 16 <= N) {                         // wave-uniform: full tile (hot path)
#pragma unroll
        for (int r = 0; r < 8; ++r) hp[r * 128] = c[r];
    } else {                                    // cold: partial last tile
#pragma unroll
        for (int r = 0; r < 8; ++r)
            if (rbase + r < N) hp[r * 128] = c[r];
    }
}

// ---------------------------------------------------------------------------
// Layer-1 attention scores: s_src/s_dst [N,8]
// ---------------------------------------------------------------------------
__global__ void scores1(const float* __restrict__ h, const float* __restrict__ asrc,
                        const float* __restrict__ adst, float* __restrict__ ssrc,
                        float* __restrict__ sdst, int N) {
    int tid = blockIdx.x * blockDim.x + threadIdx.x;
    if (tid >= N * 8) return;
    int n = tid >> 3, hh = tid & 7;
    const float* hp = h + (size_t)n * 128 + hh * 16;
    const float* ap = asrc + hh * 16;
    const float* bp = adst + hh * 16;
    float ss = 0.0f, sd = 0.0f;
#pragma unroll
    for (int c = 0; c < 16; ++c) { float v = hp[c]; ss += v * ap[c]; sd += v * bp[c]; }
    ssrc[tid] = ss;
    sdst[tid] = sd;
}

// ---------------------------------------------------------------------------
// Edge sweeps, layer 1 (8 heads). Edges i<E from edge_index; i>=E self-loops.
// ---------------------------------------------------------------------------
__global__ void edge_max1(const int* __restrict__ ei, const float* __restrict__ ssrc,
                          const float* __restrict__ sdst, unsigned* __restrict__ m1,
                          int E, int N) {
    int i = blockIdx.x * blockDim.x + threadIdx.x;
    if (i >= E + N) return;
    int s, d;
    if (i < E) { s = ei[i]; d = ei[E + i]; } else { s = d = i - E; }
#pragma unroll
    for (int hh = 0; hh < 8; ++hh) {
        float e = lrelu(ssrc[s * 8 + hh] + sdst[d * 8 + hh]);
        atomicMax(&m1[d * 8 + hh], fenc(e));
    }
}

__global__ void edge_sum1(const int* __restrict__ ei, const float* __restrict__ ssrc,
                          const float* __restrict__ sdst, const unsigned* __restrict__ m1,
                          float* __restrict__ z1, int E, int N) {
    int i = blockIdx.x * blockDim.x + threadIdx.x;
    if (i >= E + N) return;
    int s, d;
    if (i < E) { s = ei[i]; d = ei[E + i]; } else { s = d = i - E; }
#pragma unroll
    for (int hh = 0; hh < 8; ++hh) {
        float e = lrelu(ssrc[s * 8 + hh] + sdst[d * 8 + hh]);
        atomicAdd(&z1[d * 8 + hh], __expf(e - fdec(m1[d * 8 + hh])));
    }
}

// One wave per edge; lane l owns channels [4l, 4l+4) of the 128-wide row.
__global__ void edge_msg1(const int* __restrict__ ei, const float* __restrict__ ssrc,
                          const float* __restrict__ sdst, const unsigned* __restrict__ m1,
                          const float* __restrict__ z1, const float* __restrict__ h,
                          float* __restrict__ out1, int E, int N) {
    int gid  = blockIdx.x * blockDim.x + threadIdx.x;
    int edge = gid >> 5;
    int lane = threadIdx.x & 31;
    if (edge >= E + N) return;
    int s, d;
    if (edge < E) { s = ei[edge]; d = ei[E + edge]; } else { s = d = edge - E; }
    __builtin_prefetch(h + (size_t)s * 128, 0, 1);
    int hh = lane >> 2;
    float e = lrelu(ssrc[s * 8 + hh] + sdst[d * 8 + hh]);
    float alpha = __expf(e - fdec(m1[d * 8 + hh])) / (z1[d * 8 + hh] + GAT_EPS);
    const float4 hv = *(const float4*)(h + (size_t)s * 128 + lane * 4);
    float* o = out1 + (size_t)d * 128 + lane * 4;
    atomicAdd(o + 0, hv.x * alpha);
    atomicAdd(o + 1, hv.y * alpha);
    atomicAdd(o + 2, hv.z * alpha);
    atomicAdd(o + 3, hv.w * alpha);
}

// ---------------------------------------------------------------------------
// Layer-2 GEMM: h2[N,3] = relu(out1)[N,128] @ W2[128,3], WMMA f32 16x16x4,
// 32 K-steps. ReLU fused into A loads. Row/col indices are clamped for loads;
// stores: wave-uniform full-tile path with a single jl<3 exec region.
// ---------------------------------------------------------------------------
__global__ void gemm2_wmma(const float* __restrict__ out1, const float* __restrict__ W2,
                           float* __restrict__ h2, int N, int nRowTiles) {
    int gid  = blockIdx.x * blockDim.x + threadIdx.x;
    int wave = gid >> 5;
    int lane = threadIdx.x & 31;
    if (wave >= nRowTiles) return;
    int m0 = wave * 16;
    int mlane = m0 + (lane & 15);
    mlane = mlane < N ? mlane : 0;
    int klo = (lane >> 4) << 1;
    int jl  = lane & 15;
    int jc  = jl < 3 ? jl : 0;                  // clamp: cols 3..15 duplicate col 0
    const float* arow = out1 + (size_t)mlane * 128 + klo;   // 8B-aligned
    v8f c = {};
#pragma unroll
    for (int k = 0; k < 128; k += 4) {
        v2f a = *(const v2f*)(arow + k);        // one global_load_b64
        a.x = fmaxf(a.x, 0.0f);
        a.y = fmaxf(a.y, 0.0f);
        v2f b;
        b.x = W2[(k + klo) * 3 + jc];
        b.y = W2[(k + klo + 1) * 3 + jc];
        c = __builtin_amdgcn_wmma_f32_16x16x4_f32(false, a, false, b, (short)0, c,
                                                  false, false);
    }
    int rbase = m0 + ((lane >> 4) << 3);
    float* hp = h2 + (size_t)rbase * 3 + jl;
    if (m0 + 16 <= N) {                         // wave-uniform: full tile
        if (jl < 3) {                           // single divergent region, 8 stores
#pragma unroll
            for (int r = 0; r < 8; ++r) hp[r * 3] = c[r];
        }
    } else {
#pragma unroll
        for (int r = 0; r < 8; ++r)
            if (jl < 3 && rbase + r < N) hp[r * 3] = c[r];
    }
}

__global__ void scores2(const float* __restrict__ h2, const float* __restrict__ asrc,
                        const float* __restrict__ adst, float* __restrict__ ssrc,
                        float* __restrict__ sdst, int N) {
    int n = blockIdx.x * blockDim.x + threadIdx.x;
    if (n >= N) return;
    float ss = 0.0f, sd = 0.0f;
#pragma unroll
    for (int j = 0; j < 3; ++j) { float v = h2[n * 3 + j]; ss += v * asrc[j]; sd += v * adst[j]; }
    ssrc[n] = ss;
    sdst[n] = sd;
}

__global__ void edge_max2(const int* __restrict__ ei, const float* __restrict__ ssrc,
                          const float* __restrict__ sdst, unsigned* __restrict__ m2,
                          int E, int N) {
    int i = blockIdx.x * blockDim.x + threadIdx.x;
    if (i >= E + N) return;
    int s, d;
    if (i < E) { s = ei[i]; d = ei[E + i]; } else { s = d = i - E; }
    atomicMax(&m2[d], fenc(lrelu(ssrc[s] + sdst[d])));
}

__global__ void edge_sum2(const int* __restrict__ ei, const float* __restrict__ ssrc,
                          const float* __restrict__ sdst, const unsigned* __restrict__ m2,
                          float* __restrict__ z2, int E, int N) {
    int i = blockIdx.x * blockDim.x + threadIdx.x;
    if (i >= E + N) return;
    int s, d;
    if (i < E) { s = ei[i]; d = ei[E + i]; } else { s = d = i - E; }
    float e = lrelu(ssrc[s] + sdst[d]);
    atomicAdd(&z2[d], __expf(e - fdec(m2[d])));
}

__global__ void edge_msg2(const int* __restrict__ ei, const float* __restrict__ ssrc,
                          const float* __restrict__ sdst, const unsigned* __restrict__ m2,
                          const float* __restrict__ z2, const float* __restrict__ h2,
                          float* __restrict__ out, int E, int N) {
    int i = blockIdx.x * blockDim.x + threadIdx.x;
    if (i >= E + N) return;
    int s, d;
    if (i < E) { s = ei[i]; d = ei[E + i]; } else { s = d = i - E; }
    float e = lrelu(ssrc[s] + sdst[d]);
    float alpha = __expf(e - fdec(m2[d])) / (z2[d] + GAT_EPS);
#pragma unroll
    for (int j = 0; j < 3; ++j) atomicAdd(&out[d * 3 + j], h2[s * 3 + j] * alpha);
}

// ---------------------------------------------------------------------------
extern "C" void kernel_launch(void* const* d_in, const int* in_sizes, int n_in,
                              void* d_out, int out_size, void* d_ws, size_t ws_size,
                              hipStream_t stream) {
    const float* x      = (const float*)d_in[0];
    const int*   ei     = (const int*)d_in[1];
    // d_in[2] = edge_type (unused by reference)
    const float* W1     = (const float*)d_in[3];
    const float* a_src1 = (const float*)d_in[4];
    const float* a_dst1 = (const float*)d_in[5];
    const float* b1     = (const float*)d_in[6];
    const float* W2     = (const float*)d_in[7];
    const float* a_src2 = (const float*)d_in[8];
    const float* a_dst2 = (const float*)d_in[9];
    const float* b2     = (const float*)d_in[10];
    float* out = (float*)d_out;

    int N  = in_sizes[0] / 16;
    int E  = in_sizes[1] / 2;
    int E2 = E + N;

    float* ws = (float*)d_ws;
    float*    h1    = ws;               ws += (size_t)N * 128;
    float*    out1  = ws;               ws += (size_t)N * 128;
    float*    ssrc1 = ws;               ws += (size_t)N * 8;
    float*    sdst1 = ws;               ws += (size_t)N * 8;
    unsigned* m1    = (unsigned*)ws;    ws += (size_t)N * 8;
    float*    z1    = ws;               ws += (size_t)N * 8;
    float*    h2    = ws;               ws += (size_t)N * 3;
    float*    ssrc2 = ws;               ws += (size_t)N;
    float*    sdst2 = ws;               ws += (size_t)N;
    unsigned* m2    = (unsigned*)ws;    ws += (size_t)N;
    float*    z2    = ws;

    const int B = 256;
    int nRowTiles = (N + 15) / 16;

    gat_init<<<(N * 128 + B - 1) / B, B, 0, stream>>>(out1, m1, z1, out, m2, z2, b1, b2, N);

    // Layer 1
    {
        int waves = nRowTiles * 8;                       // 8 column tiles
        gemm1_wmma<<<(waves + 7) / 8, B, 0, stream>>>(x, W1, h1, N, nRowTiles);
    }
    scores1<<<(N * 8 + B - 1) / B, B, 0, stream>>>(h1, a_src1, a_dst1, ssrc1, sdst1, N);
    edge_max1<<<(E2 + B - 1) / B, B, 0, stream>>>(ei, ssrc1, sdst1, m1, E, N);
    edge_sum1<<<(E2 + B - 1) / B, B, 0, stream>>>(ei, ssrc1, sdst1, m1, z1, E, N);
    edge_msg1<<<(E2 + 7) / 8, B, 0, stream>>>(ei, ssrc1, sdst1, m1, z1, h1, out1, E, N);

    // Layer 2
    gemm2_wmma<<<(nRowTiles + 7) / 8, B, 0, stream>>>(out1, W2, h2, N, nRowTiles);
    scores2<<<(N + B - 1) / B, B, 0, stream>>>(h2, a_src2, a_dst2, ssrc2, sdst2, N);
    edge_max2<<<(E2 + B - 1) / B, B, 0, stream>>>(ei, ssrc2, sdst2, m2, E, N);
    edge_sum2<<<(E2 + B - 1) / B, B, 0, stream>>>(ei, ssrc2, sdst2, m2, z2, E, N);
    edge_msg2<<<(E2 + B - 1) / B, B, 0, stream>>>(ei, ssrc2, sdst2, m2, z2, h2, out, E, N);
}